// Bipartite_44014824849868
// MI455X (gfx1250) — compile-verified
//
#include <hip/hip_runtime.h>
#include <float.h>

// ---------------------------------------------------------------------------
// Bipartite attention policy:
//   s_task = task_nf @ w_att[:D]; s_ag = ag_nf @ w_att[D:]; ag_lin = ag_nf @ w_ag
//   policy[i,j]  = softmax_j( leaky(s_ag[i] + s_task[j]) masked by finished )
//   ag_policy[i] = softmax_i( leaky(ag_lin[i]) )
// Dominant cost: 134MB policy write -> stream it, recompute exp, never
// materialize scores. Matvecs done with V_WMMA_F32_16X16X4_F32 (exact f32).
// ---------------------------------------------------------------------------

typedef float  v2f __attribute__((ext_vector_type(2)));
typedef float  v8f __attribute__((ext_vector_type(8)));

#define NEG_SLOPE 0.01f

__device__ __forceinline__ float lrelu(float x) { return x > 0.0f ? x : NEG_SLOPE * x; }

// ---------------------------------------------------------------------------
// Kernel 1: all three matvecs via WMMA f32 16x16x4.
// One wave (32 threads) per 16-row tile. A is pre-multiplied elementwise by
// the weight vector (so the contraction value is layout-permutation
// invariant); B is all-ones (layout invariant). C[m][n] = dot(row m, w) for
// every column n. Extract rows 0-7 from lane 0 (c[0..7]) and rows 8-15 from
// lane 16 per the documented C/D layout.
// ---------------------------------------------------------------------------
__global__ __launch_bounds__(32)
void dot_wmma_kernel(const float* __restrict__ nf,
                     const float* __restrict__ w_att,   // [2*D]
                     const float* __restrict__ w_ag,    // [D]
                     const int*   __restrict__ ag_idx,
                     const int*   __restrict__ task_idx,
                     float* __restrict__ s_task,
                     float* __restrict__ s_ag,
                     float* __restrict__ ag_lin,
                     int n_ag_tiles, int D)
{
    const int tile = blockIdx.x;
    const int lane = threadIdx.x;          // 0..31, wave32
    const bool hi  = lane >= 16;
    const int  m   = lane & 15;            // row within tile
    const v2f  bone = {1.0f, 1.0f};        // all-ones B (layout invariant)

    if (tile < n_ag_tiles) {
        // ----- agent tile: two dots (w_att[D:] and w_ag) -----
        const int row = ag_idx[tile * 16 + m];
        const float* __restrict__ a  = nf + (long long)row * D;
        const float* __restrict__ wd = w_att + D;
        v8f c1 = {}; v8f c2 = {};
        for (int kb = 0; kb < D; kb += 4) {
            // documented f32 A layout: V0 = K0(lanes0-15)/K2(lanes16-31), V1 = K1/K3
            const int k0 = kb + (hi ? 2 : 0);
            const float a0 = a[k0], a1 = a[k0 + 1];
            v2f A1 = { a0 * wd[k0],   a1 * wd[k0 + 1] };
            v2f A2 = { a0 * w_ag[k0], a1 * w_ag[k0 + 1] };
            c1 = __builtin_amdgcn_wmma_f32_16x16x4_f32(false, A1, false, bone,
                                                       (short)0, c1, false, false);
            c2 = __builtin_amdgcn_wmma_f32_16x16x4_f32(false, A2, false, bone,
                                                       (short)0, c2, false, false);
        }
        if (m == 0) {  // lane 0 -> rows 0..7, lane 16 -> rows 8..15
            const int base = tile * 16 + (hi ? 8 : 0);
            #pragma unroll
            for (int r = 0; r < 8; ++r) {
                s_ag[base + r]   = c1[r];
                ag_lin[base + r] = c2[r];
            }
        }
    } else {
        // ----- task tile: one dot (w_att[:D]) -----
        const int t   = tile - n_ag_tiles;
        const int row = task_idx[t * 16 + m];
        const float* __restrict__ a  = nf + (long long)row * D;
        const float* __restrict__ ws = w_att;
        v8f c = {};
        for (int kb = 0; kb < D; kb += 4) {
            const int k0 = kb + (hi ? 2 : 0);
            v2f A = { a[k0] * ws[k0], a[k0 + 1] * ws[k0 + 1] };
            c = __builtin_amdgcn_wmma_f32_16x16x4_f32(false, A, false, bone,
                                                      (short)0, c, false, false);
        }
        if (m == 0) {
            const int base = t * 16 + (hi ? 8 : 0);
            #pragma unroll
            for (int r = 0; r < 8; ++r) s_task[base + r] = c[r];
        }
    }
}

// ---------------------------------------------------------------------------
// Kernel 2: global max over unfinished s_task (monotonic leaky => row max is
// leaky(s_ag[i] + T) exactly).
// ---------------------------------------------------------------------------
__global__ __launch_bounds__(1024)
void tmax_kernel(const float* __restrict__ s_task,
                 const unsigned char* __restrict__ finished,
                 int n_task, float* __restrict__ Tout)
{
    __shared__ float red[1024];
    const int tid = threadIdx.x;
    float mx = -FLT_MAX;
    for (int j = tid; j < n_task; j += 1024)
        if (!finished[j]) mx = fmaxf(mx, s_task[j]);
    red[tid] = mx;
    __syncthreads();
    for (int s = 512; s > 0; s >>= 1) {
        if (tid < s) red[tid] = fmaxf(red[tid], red[tid + s]);
        __syncthreads();
    }
    if (tid == 0) Tout[0] = red[0];
}

// ---------------------------------------------------------------------------
// Kernel 3: fused row-softmax + streaming write of policy.
// 8 agents per block: s_task is loaded once per iteration and reused for all
// 8 rows (cuts L2 traffic ~8x); writes are fully coalesced per-row streams.
// Finished tasks: score = -FLT_MAX (== jnp.finfo.min) -> exp == 0 exactly.
// ---------------------------------------------------------------------------
#define AGB 8
__global__ __launch_bounds__(256)
void policy_kernel(const float* __restrict__ s_task,
                   const float* __restrict__ s_ag,
                   const unsigned char* __restrict__ finished,
                   const float* __restrict__ Tptr,
                   float* __restrict__ out,
                   int n_ag, int n_task)
{
    const int tid   = threadIdx.x;
    const int abase = blockIdx.x * AGB;
    const float T   = Tptr[0];

    float a[AGB], m[AGB], acc[AGB];
    #pragma unroll
    for (int q = 0; q < AGB; ++q) {
        a[q]   = (abase + q < n_ag) ? s_ag[abase + q] : 0.0f;
        m[q]   = lrelu(a[q] + T);
        acc[q] = 0.0f;
    }

    // pass 1: per-agent exp-sums
    for (int j = tid; j < n_task; j += 256) {
        const float e = finished[j] ? -FLT_MAX : s_task[j];
        #pragma unroll
        for (int q = 0; q < AGB; ++q)
            acc[q] += expf(lrelu(a[q] + e) - m[q]);
    }

    __shared__ float red[256];
    __shared__ float inv_sh[AGB];
    for (int q = 0; q < AGB; ++q) {
        red[tid] = acc[q];
        __syncthreads();
        for (int s = 128; s > 0; s >>= 1) {
            if (tid < s) red[tid] += red[tid + s];
            __syncthreads();
        }
        if (tid == 0) inv_sh[q] = 1.0f / red[0];
        __syncthreads();
    }
    float iv[AGB];
    #pragma unroll
    for (int q = 0; q < AGB; ++q) iv[q] = inv_sh[q];

    // pass 2: streaming write (dominant HBM cost: 134MB total)
    for (int j = tid; j < n_task; j += 256) {
        const float e = finished[j] ? -FLT_MAX : s_task[j];
        #pragma unroll
        for (int q = 0; q < AGB; ++q) {
            if (abase + q < n_ag)
                out[(size_t)(abase + q) * n_task + j] =
                    expf(lrelu(a[q] + e) - m[q]) * iv[q];
        }
    }
}

// ---------------------------------------------------------------------------
// Kernel 4: ag_policy = softmax(leaky(ag_lin)) over the 2048 agents.
// ---------------------------------------------------------------------------
__global__ __launch_bounds__(1024)
void ag_policy_kernel(const float* __restrict__ ag_lin,
                      float* __restrict__ out, int n_ag)
{
    __shared__ float red[1024];
    const int tid = threadIdx.x;

    float mx = -FLT_MAX;
    for (int i = tid; i < n_ag; i += 1024) mx = fmaxf(mx, lrelu(ag_lin[i]));
    red[tid] = mx;
    __syncthreads();
    for (int s = 512; s > 0; s >>= 1) {
        if (tid < s) red[tid] = fmaxf(red[tid], red[tid + s]);
        __syncthreads();
    }
    const float M = red[0];
    __syncthreads();

    float sum = 0.0f;
    for (int i = tid; i < n_ag; i += 1024) sum += expf(lrelu(ag_lin[i]) - M);
    red[tid] = sum;
    __syncthreads();
    for (int s = 512; s > 0; s >>= 1) {
        if (tid < s) red[tid] += red[tid + s];
        __syncthreads();
    }
    const float inv = 1.0f / red[0];

    for (int i = tid; i < n_ag; i += 1024)
        out[i] = expf(lrelu(ag_lin[i]) - M) * inv;
}

// ---------------------------------------------------------------------------
extern "C" void kernel_launch(void* const* d_in, const int* in_sizes, int n_in,
                              void* d_out, int out_size, void* d_ws, size_t ws_size,
                              hipStream_t stream) {
    const float*         nf       = (const float*)d_in[0];
    const float*         w_att    = (const float*)d_in[1];   // [2*D]
    const float*         w_ag     = (const float*)d_in[2];   // [D]
    const int*           ag_idx   = (const int*)d_in[3];
    const int*           task_idx = (const int*)d_in[4];
    const unsigned char* finished = (const unsigned char*)d_in[5];

    const int D      = in_sizes[2];      // 128
    const int n_ag   = in_sizes[3];      // 2048
    const int n_task = in_sizes[4];      // 16384

    float* out_policy = (float*)d_out;                               // [n_ag*n_task]
    float* out_agpol  = out_policy + (size_t)n_ag * n_task;          // [n_ag]

    float* ws      = (float*)d_ws;
    float* s_task  = ws;                  // [n_task]
    float* s_ag    = s_task + n_task;     // [n_ag]
    float* ag_lin  = s_ag + n_ag;         // [n_ag]
    float* Tbuf    = ag_lin + n_ag;       // [1]

    const int ag_tiles   = n_ag / 16;
    const int task_tiles = n_task / 16;

    dot_wmma_kernel<<<ag_tiles + task_tiles, 32, 0, stream>>>(
        nf, w_att, w_ag, ag_idx, task_idx, s_task, s_ag, ag_lin, ag_tiles, D);

    tmax_kernel<<<1, 1024, 0, stream>>>(s_task, finished, n_task, Tbuf);

    policy_kernel<<<(n_ag + AGB - 1) / AGB, 256, 0, stream>>>(
        s_task, s_ag, finished, Tbuf, out_policy, n_ag, n_task);

    ag_policy_kernel<<<1, 1024, 0, stream>>>(ag_lin, out_agpol, n_ag);
}